// MPNN2_91122026152488
// MI455X (gfx1250) — compile-verified
//
#include <hip/hip_runtime.h>
#include <hip/hip_bf16.h>

typedef float v8f __attribute__((ext_vector_type(8)));
typedef float v2f __attribute__((ext_vector_type(2)));

#define DIMC 32
#define HC   32
#define FN   16
#define FE   8
#define NG   64
#define BIGN 320   // 256 (W cols, [f*32+h]) + 32 (nn_b fold) + 32 (root_w fold)

// ---------------- Kernel 0: zero accumulators + build BIG [32 x 320] ----------------
// BIG[d, f*32+h]   = nn_w[(d*32+h)*8 + f]   (cols 0..255)
// BIG[d, 256+h]    = nn_b[d*32+h]           (cols 256..287)
// BIG[d, 288+h]    = root_w[d*32+h]         (cols 288..319)
__global__ void mpnn_prep(const float* __restrict__ nn_w, const float* __restrict__ nn_b,
                          const float* __restrict__ root_w,
                          float* __restrict__ BIG, float* __restrict__ agg,
                          float* __restrict__ u, int N) {
    int t = blockIdx.x * blockDim.x + threadIdx.x;
    int stride = gridDim.x * blockDim.x;
    for (int i = t; i < N * HC; i += stride) agg[i] = 0.0f;
    if (t < NG * HC) u[t] = 0.0f;
    for (int i = t; i < DIMC * BIGN; i += stride) {
        int d = i / BIGN, c = i % BIGN;
        float v;
        if (c < 256)      { int f = c >> 5, h = c & 31; v = nn_w[(d * HC + h) * FE + f]; }
        else if (c < 288) { v = nn_b[d * HC + (c - 256)]; }
        else              { v = root_w[d * HC + (c - 288)]; }
        BIG[i] = v;
    }
}

// ---------------- Kernel 1: out = relu(x @ lin0_w^T + lin0_b)  [N x 32] ----------------
__global__ void mpnn_lin0(const float* __restrict__ x, const float* __restrict__ w,
                          const float* __restrict__ b, float* __restrict__ out, int N) {
    int t = blockIdx.x * blockDim.x + threadIdx.x;
    int n = t >> 5, d = t & 31;
    if (n >= N) return;                       // wave-uniform
    float s = b[d];
    #pragma unroll
    for (int f = 0; f < FN; ++f) s += x[n * FN + f] * w[d * FN + f];
    out[n * DIMC + d] = fmaxf(s, 0.0f);
}

// ---------------- Kernel 2: P = out @ BIG   ([N,32] x [32,320])  via WMMA f32 16x16x4 ----
// One wave per 16x16 output tile; K=32 -> 8 WMMA ops per tile.
// A frag (16x4): lanes 0-15 -> K=k,k+1 ; lanes 16-31 -> K=k+2,k+3 (per ISA layout).
// B frag (4x16): mirrored K split, N striped over lanes 0-15 / 16-31.
// D (16x16 f32): VGPR v -> M = v + (lane<16 ? 0 : 8), N = lane&15.
__global__ void mpnn_gemm_wmma(const float* __restrict__ out, const float* __restrict__ BIG,
                               float* __restrict__ P, int N, int mtiles) {
    int wave = (blockIdx.x * blockDim.x + threadIdx.x) >> 5;
    int lane = threadIdx.x & 31;
    const int ntiles = BIGN / 16;             // 20
    int mtile = wave / ntiles;
    int ntile = wave % ntiles;
    if (mtile >= mtiles) return;              // wave-uniform exit
    int mbase = mtile * 16;
    int nbase = ntile * 16;
    int hi    = lane >> 4;                    // 0: lanes 0-15, 1: lanes 16-31
    int l15   = lane & 15;
    int arow  = mbase + l15;
    int arowc = arow < N ? arow : N - 1;      // clamp loads -> EXEC stays all-1s

    v8f acc = {};
    #pragma unroll
    for (int k = 0; k < DIMC; k += 4) {
        int klo = k + (hi ? 2 : 0);
        v2f a, b;
        a.x = out[arowc * DIMC + klo];
        a.y = out[arowc * DIMC + klo + 1];
        b.x = BIG[klo * BIGN + nbase + l15];
        b.y = BIG[(klo + 1) * BIGN + nbase + l15];
        acc = __builtin_amdgcn_wmma_f32_16x16x4_f32(
            /*neg_a=*/false, a, /*neg_b=*/false, b,
            /*c_mod=*/(short)0, acc, /*reuse_a=*/false, /*reuse_b=*/false);
    }

    float* pout = P + (size_t)(mbase + hi * 8) * BIGN + nbase + l15;
    if (mbase + 16 <= N) {
        // full tile: straight, unpredicated stores (wave-uniform fast path)
        #pragma unroll
        for (int v = 0; v < 8; ++v) pout[(size_t)v * BIGN] = acc[v];
    } else {
        // single ragged tail tile: per-element predication (after all WMMAs)
        #pragma unroll
        for (int v = 0; v < 8; ++v) {
            int m = mbase + v + hi * 8;
            if (m < N) P[(size_t)m * BIGN + nbase + l15] = acc[v];
        }
    }
}

// ---------------- Kernel 3: edge messages + scatter-add --------------------------------
// msg[e,h] = P[src,256+h] + sum_f ea[e,f] * P[src, f*32+h];  agg[dst,h] += msg
__global__ void mpnn_edges(const int* __restrict__ ei, const float* __restrict__ ea,
                           const float* __restrict__ P, float* __restrict__ agg, int E) {
    int t = blockIdx.x * blockDim.x + threadIdx.x;
    int e = t >> 5, h = t & 31;
    if (e >= E) return;                       // wave-uniform
    int src = ei[e];
    int dst = ei[E + e];
    const float* pr = P + (size_t)src * BIGN;
    float acc = pr[256 + h];
    float ef = (h < FE) ? ea[(size_t)e * FE + h] : 0.0f;   // 8 lanes hold edge_attr
    #pragma unroll
    for (int f = 0; f < FE; ++f) {
        float c = __shfl(ef, f, 32);
        acc += c * pr[f * HC + h];
    }
    atomicAdd(&agg[(size_t)dst * HC + h], acc);
}

// ---------------- Kernel 4: node combine + global-add pool -----------------------------
// hgt[n,h] = relu(agg[n,h] + P[n,288+h] + conv_b[h]);  u[batch[n],h] += hgt
__global__ void mpnn_combine(const float* __restrict__ agg, const float* __restrict__ P,
                             const float* __restrict__ conv_b, const int* __restrict__ batch,
                             float* __restrict__ u, int N) {
    int t = blockIdx.x * blockDim.x + threadIdx.x;
    int n = t >> 5, h = t & 31;
    if (n >= N) return;                       // wave-uniform
    float hv = agg[(size_t)n * HC + h] + P[(size_t)n * BIGN + 288 + h] + conv_b[h];
    hv = fmaxf(hv, 0.0f);
    atomicAdd(&u[(size_t)batch[n] * HC + h], hv);
}

// ---------------- Kernel 5: MLP head ---------------------------------------------------
__global__ void mpnn_head(const float* __restrict__ u, const float* __restrict__ w1,
                          const float* __restrict__ b1, const float* __restrict__ w2,
                          const float* __restrict__ b2, float* __restrict__ o) {
    int g = threadIdx.x;
    if (g >= NG) return;
    float acc = b2[0];
    #pragma unroll
    for (int j = 0; j < 16; ++j) {
        float s = b1[j];
        #pragma unroll
        for (int h = 0; h < HC; ++h) s += u[g * HC + h] * w1[j * HC + h];
        acc += fmaxf(s, 0.0f) * w2[j];
    }
    o[g] = acc;
}

extern "C" void kernel_launch(void* const* d_in, const int* in_sizes, int n_in,
                              void* d_out, int out_size, void* d_ws, size_t ws_size,
                              hipStream_t stream) {
    const float* x       = (const float*)d_in[0];
    const int*   ei      = (const int*)  d_in[1];
    const float* ea      = (const float*)d_in[2];
    const int*   batch   = (const int*)  d_in[3];
    const float* lin0_w  = (const float*)d_in[4];
    const float* lin0_b  = (const float*)d_in[5];
    const float* nn_w    = (const float*)d_in[6];
    const float* nn_b    = (const float*)d_in[7];
    const float* root_w  = (const float*)d_in[8];
    const float* conv_b  = (const float*)d_in[9];
    const float* lin1_w  = (const float*)d_in[10];
    const float* lin1_b  = (const float*)d_in[11];
    const float* lin2_w  = (const float*)d_in[12];
    const float* lin2_b  = (const float*)d_in[13];
    float* o = (float*)d_out;

    const int N = in_sizes[0] / FN;       // 12500
    const int E = in_sizes[2] / FE;       // 200000

    // workspace layout (floats)
    float* ws   = (float*)d_ws;
    float* outN = ws;                                 // N*32
    float* BIG  = outN + (size_t)N * DIMC;            // 32*320
    float* P    = BIG  + (size_t)DIMC * BIGN;         // N*320
    float* agg  = P    + (size_t)N * BIGN;            // N*32
    float* u    = agg  + (size_t)N * HC;              // 64*32

    // 0: zero agg/u, build BIG
    mpnn_prep<<<512, 256, 0, stream>>>(nn_w, nn_b, root_w, BIG, agg, u, N);

    // 1: lin0 + relu
    {
        int threads = N * HC;
        mpnn_lin0<<<(threads + 255) / 256, 256, 0, stream>>>(x, lin0_w, lin0_b, outN, N);
    }

    // 2: WMMA GEMM  P = outN @ BIG
    {
        int mtiles = (N + 15) / 16;
        int waves  = mtiles * (BIGN / 16);
        mpnn_gemm_wmma<<<(waves + 7) / 8, 256, 0, stream>>>(outN, BIG, P, N, mtiles);
    }

    // 3: edge messages + scatter
    {
        long long threads = (long long)E * HC;
        mpnn_edges<<<(int)((threads + 255) / 256), 256, 0, stream>>>(ei, ea, P, agg, E);
    }

    // 4: combine + pool
    {
        int threads = N * HC;
        mpnn_combine<<<(threads + 255) / 256, 256, 0, stream>>>(agg, P, conv_b, batch, u, N);
    }

    // 5: head
    mpnn_head<<<1, 64, 0, stream>>>(u, lin1_w, lin1_b, lin2_w, lin2_b, o);
}